// deform_dss_cifar_48996986913003
// MI455X (gfx1250) — compile-verified
//
#include <hip/hip_runtime.h>
#include <hip/hip_bf16.h>

typedef __attribute__((ext_vector_type(2))) float v2f;
typedef __attribute__((ext_vector_type(8))) float v8f;

#define BSZ 16
#define SSZ 4

// ---------------------------------------------------------------------------
// weight transpose: wgt (Cout, Cin, 9) -> wT[m][k][ci_pad] zero-padded.
// Makes the WMMA A-fragment a contiguous aligned b64 load.
// ---------------------------------------------------------------------------
__global__ void wtrans_kernel(const float* __restrict__ w, float* __restrict__ wT,
                              int Cout, int Cin, int Cinp) {
  int total = Cout * 9 * Cinp;
  int i = blockIdx.x * 256 + threadIdx.x;
  if (i >= total) return;
  int ci = i % Cinp; int t = i / Cinp;
  int k = t % 9; int m = t / 9;
  wT[i] = (ci < Cin) ? w[(m * Cin + ci) * 9 + k] : 0.0f;
}

// ---------------------------------------------------------------------------
// fused im2col + WMMA fp32 3x3 conv (pad 1), tap-major reduction.
// Block = 128 thr (4 waves) sharing one 16-pixel tile; wave w owns Cout tile
// blockIdx.y*4+w. Patch chunks (16ch x 16px) staged in LDS (branchless fill).
// PADDED=true only for conv1 (Cin 3 zero-padded to 16).
// ---------------------------------------------------------------------------
template <bool PADDED>
__global__ void conv3x3_wmma_kernel(const float* __restrict__ in,
                                    const float* __restrict__ wT,
                                    const float* __restrict__ bias,
                                    float* __restrict__ out,
                                    int Cin_real, int Cinp, int Cout, int H, int W,
                                    int in_b_stride, int in_c_stride,
                                    int do_relu) {
  __shared__ float P[256];                       // [cg][n]
  const int HW = H * W;
  const int ctiles = Cout >> 4;
  const int tid = threadIdx.x;
  const int wave = tid >> 5, lane = tid & 31;
  const int nl = lane & 15, half = lane >> 4;
  const int pt = blockIdx.x;
  int ct = blockIdx.y * 4 + wave;
  const bool ct_valid = (ct < ctiles);
  if (!ct_valid) ct = ctiles - 1;                // clamp: loads stay safe, store skipped
  const int b = blockIdx.z;
  const int m = ct * 16 + nl;
  const int Rpad = 9 * Cinp;
  const float* inb = in + (long)b * in_b_stride;

  // fill-phase mapping: 16 pixels x 8 channel-pairs
  const int fn = tid & 15;
  const int fcp = tid >> 4;
  const int fp = pt * 16 + fn;
  const int fy = fp / W, fx = fp % W;

  v8f acc = {};
  const int gblocks = Cinp >> 4;
  for (int k = 0; k < 9; ++k) {
    const int dy = k / 3 - 1;
    const int dx = k % 3 - 1;
    for (int gb = 0; gb < gblocks; ++gb) {
      // ---- cooperative branchless patch fill ----
      {
        int yy = fy + dy, xx = fx + dx;
        float ms = ((yy >= 0) && (yy < H) && (xx >= 0) && (xx < W)) ? 1.0f : 0.0f;
        int idx = min(max(yy, 0), H - 1) * W + min(max(xx, 0), W - 1);
        int ci0 = gb * 16 + fcp * 2;
        float v0, v1;
        if (PADDED) {
          int c0 = min(ci0, Cin_real - 1);
          int c1 = min(ci0 + 1, Cin_real - 1);
          float m0 = (ci0 < Cin_real) ? ms : 0.0f;
          float m1 = (ci0 + 1 < Cin_real) ? ms : 0.0f;
          v0 = inb[c0 * in_c_stride + idx] * m0;
          v1 = inb[c1 * in_c_stride + idx] * m1;
        } else {
          const float* q = inb + ci0 * in_c_stride + idx;
          v0 = q[0] * ms;
          v1 = q[in_c_stride] * ms;
        }
        __syncthreads();                         // protect prior chunk reads
        P[(fcp * 2) * 16 + fn] = v0;
        P[(fcp * 2 + 1) * 16 + fn] = v1;
        __syncthreads();
      }
      // ---- 4 WMMA steps over the 16-channel chunk ----
      const float* wrow = wT + m * Rpad + k * Cinp + gb * 16;
      for (int cc = 0; cc < 4; ++cc) {
        int cg = cc * 4 + half * 2;
        v2f av = *(const v2f*)(wrow + cg);       // aligned b64
        v2f bv;
        bv.x = P[cg * 16 + nl];
        bv.y = P[(cg + 1) * 16 + nl];
        acc = __builtin_amdgcn_wmma_f32_16x16x4_f32(false, av, false, bv,
                                                    (short)0, acc, false, false);
      }
    }
  }
  if (ct_valid) {
    const int col = pt * 16 + nl;
    for (int v = 0; v < 8; ++v) {
      int row = ct * 16 + v + half * 8;
      float val = acc[v] + bias[row];
      if (do_relu) val = fmaxf(val, 0.0f);
      out[(long)((b * Cout + row) * HW + col)] = val;
    }
  }
}

// ---------------------------------------------------------------------------
// fused deformable-conv WMMA, tap-major. Cin==128, 8 offset groups of 16 ch.
// Bilinear corner weights/indices computed once per (k,g) chunk; sampled
// 16x16 patch chunk staged in LDS and shared by all 4 waves.
// ---------------------------------------------------------------------------
__global__ void deform_wmma_kernel(const float* __restrict__ in,
                                   const float* __restrict__ off,
                                   const float* __restrict__ wT,
                                   const float* __restrict__ bias,
                                   float* __restrict__ out,
                                   int Cin, int Cout, int H, int W, int do_relu) {
  __shared__ float P[256];
  const int HW = H * W;
  const int ctiles = Cout >> 4;
  const int tid = threadIdx.x;
  const int wave = tid >> 5, lane = tid & 31;
  const int nl = lane & 15, half = lane >> 4;
  const int pt = blockIdx.x;
  int ct = blockIdx.y * 4 + wave;
  const bool ct_valid = (ct < ctiles);
  if (!ct_valid) ct = ctiles - 1;
  const int b = blockIdx.z;
  const int m = ct * 16 + nl;
  const int Rpad = 9 * Cin;
  const float* inb = in + (long)b * Cin * HW;
  const float* offb = off + (long)b * 144 * HW;

  const int fn = tid & 15;
  const int fcp = tid >> 4;
  const int fp = pt * 16 + fn;
  const int fy = fp / W, fx = fp % W;

  v8f acc = {};
  const int groups = Cin >> 4;
  for (int k = 0; k < 9; ++k) {
    const int dy = k / 3 - 1;
    const int dx = k % 3 - 1;
    for (int g = 0; g < groups; ++g) {
      // ---- cooperative bilinear patch fill ----
      {
        int oc = (g * 9 + k) * 2;
        float ody = offb[oc * HW + fp];
        float odx = offb[(oc + 1) * HW + fp];
        float py = (float)(fy + dy) + ody;
        float px = (float)(fx + dx) + odx;
        float y0f = floorf(py), x0f = floorf(px);
        int y0 = (int)y0f, x0 = (int)x0f;
        float ly = py - y0f, lx = px - x0f;
        float w00 = (1.0f - ly) * (1.0f - lx);
        float w01 = (1.0f - ly) * lx;
        float w10 = ly * (1.0f - lx);
        float w11 = ly * lx;
        float vy0 = ((y0 >= 0) && (y0 < H)) ? 1.0f : 0.0f;
        float vy1 = ((y0 + 1 >= 0) && (y0 + 1 < H)) ? 1.0f : 0.0f;
        float vx0 = ((x0 >= 0) && (x0 < W)) ? 1.0f : 0.0f;
        float vx1 = ((x0 + 1 >= 0) && (x0 + 1 < W)) ? 1.0f : 0.0f;
        w00 *= vy0 * vx0;
        w01 *= vy0 * vx1;
        w10 *= vy1 * vx0;
        w11 *= vy1 * vx1;
        int y0c = min(max(y0, 0), H - 1);
        int y1c = min(max(y0 + 1, 0), H - 1);
        int x0c = min(max(x0, 0), W - 1);
        int x1c = min(max(x0 + 1, 0), W - 1);
        int i00 = y0c * W + x0c, i01 = y0c * W + x1c;
        int i10 = y1c * W + x0c, i11 = y1c * W + x1c;
        const float* ch0 = inb + (g * 16 + fcp * 2) * HW;
        const float* ch1 = ch0 + HW;
        float v0 = fmaf(w00, ch0[i00], fmaf(w01, ch0[i01],
                   fmaf(w10, ch0[i10], w11 * ch0[i11])));
        float v1 = fmaf(w00, ch1[i00], fmaf(w01, ch1[i01],
                   fmaf(w10, ch1[i10], w11 * ch1[i11])));
        __syncthreads();
        P[(fcp * 2) * 16 + fn] = v0;
        P[(fcp * 2 + 1) * 16 + fn] = v1;
        __syncthreads();
      }
      // ---- 4 WMMA steps ----
      const float* wrow = wT + m * Rpad + k * Cin + g * 16;
      for (int cc = 0; cc < 4; ++cc) {
        int cg = cc * 4 + half * 2;
        v2f av = *(const v2f*)(wrow + cg);
        v2f bv;
        bv.x = P[cg * 16 + nl];
        bv.y = P[(cg + 1) * 16 + nl];
        acc = __builtin_amdgcn_wmma_f32_16x16x4_f32(false, av, false, bv,
                                                    (short)0, acc, false, false);
      }
    }
  }
  if (ct_valid) {
    const int col = pt * 16 + nl;
    for (int v = 0; v < 8; ++v) {
      int row = ct * 16 + v + half * 8;
      float val = acc[v] + bias[row];
      if (do_relu) val = fmaxf(val, 0.0f);
      out[(long)((b * Cout + row) * HW + col)] = val;
    }
  }
}

// ---------------------------------------------------------------------------
// per-(channel, slice) batch-norm stats over (B, HW); biased variance.
// ---------------------------------------------------------------------------
__global__ void bn_stats_kernel(const float* __restrict__ h,
                                float* __restrict__ mean, float* __restrict__ inv,
                                int C, int HW) {
  __shared__ float ssum[256];
  __shared__ float ssq[256];
  int c = blockIdx.x;
  float s = 0.0f, sq = 0.0f;
  for (int b = 0; b < BSZ; ++b) {
    const float* row = h + (long)(b * C + c) * HW;
    for (int p = threadIdx.x; p < HW; p += 256) {
      float v = row[p];
      s += v; sq += v * v;
    }
  }
  ssum[threadIdx.x] = s; ssq[threadIdx.x] = sq;
  __syncthreads();
  for (int st = 128; st > 0; st >>= 1) {
    if (threadIdx.x < st) {
      ssum[threadIdx.x] += ssum[threadIdx.x + st];
      ssq[threadIdx.x] += ssq[threadIdx.x + st];
    }
    __syncthreads();
  }
  if (threadIdx.x == 0) {
    float n = (float)(BSZ * HW);
    float mm = ssum[0] / n;
    float var = ssq[0] / n - mm * mm;
    mean[c] = mm;
    inv[c] = rsqrtf(var + 1e-5f);
  }
}

__global__ void bn_apply_kernel(float* __restrict__ h,
                                const float* __restrict__ mean,
                                const float* __restrict__ inv,
                                const float* __restrict__ gamma,
                                const float* __restrict__ beta,
                                int cmask, int hw_shift, int total, int per_channel) {
  int i = blockIdx.x * 256 + threadIdx.x;
  if (i >= total) return;
  int c = (i >> hw_shift) & cmask;
  float g = per_channel ? gamma[c] : gamma[0];
  float be = per_channel ? beta[c] : beta[0];
  h[i] = g * (h[i] - mean[c]) * inv[c] + be;
}

// ---------------------------------------------------------------------------
// 2x2 maxpool; dims are powers of two so decode with shifts/masks.
// ---------------------------------------------------------------------------
__global__ void maxpool_kernel(const float* __restrict__ in, float* __restrict__ out,
                               int C, int H, int W, int wo_shift, int ho_shift) {
  int Wo = W >> 1, Ho = H >> 1;
  int total = BSZ * C * Ho * Wo;
  int i = blockIdx.x * 256 + threadIdx.x;
  if (i >= total) return;
  int xo = i & (Wo - 1); int t = i >> wo_shift;
  int yo = t & (Ho - 1); int bc = t >> ho_shift;
  const float* q = in + (long)bc * H * W;
  float v = fmaxf(fmaxf(q[(2 * yo) * W + 2 * xo], q[(2 * yo) * W + 2 * xo + 1]),
                  fmaxf(q[(2 * yo + 1) * W + 2 * xo], q[(2 * yo + 1) * W + 2 * xo + 1]));
  out[i] = v;
}

// same, scattering into d_out embedding with reference (B,C,S,8,8) layout
__global__ void maxpool_emb_kernel(const float* __restrict__ in, float* __restrict__ emb,
                                   int C, int H, int W, int s) {
  int Wo = W >> 1, Ho = H >> 1;                  // 8, 8
  int total = BSZ * C * Ho * Wo;
  int i = blockIdx.x * 256 + threadIdx.x;
  if (i >= total) return;
  int xo = i & (Wo - 1); int t = i >> 3;
  int yo = t & (Ho - 1); int bc = t >> 3;
  const float* q = in + (long)bc * H * W;
  float v = fmaxf(fmaxf(q[(2 * yo) * W + 2 * xo], q[(2 * yo) * W + 2 * xo + 1]),
                  fmaxf(q[(2 * yo + 1) * W + 2 * xo], q[(2 * yo + 1) * W + 2 * xo + 1]));
  emb[(long)(bc * SSZ + s) * (Ho * Wo) + yo * Wo + xo] = v;
}

// strength = relu(max over S) : emb (B,C,S,8,8) -> str (B, C*64)
__global__ void strength_kernel(const float* __restrict__ emb, float* __restrict__ str,
                                int C) {
  int total = BSZ * C * 64;
  int i = blockIdx.x * 256 + threadIdx.x;
  if (i >= total) return;
  int q = i & 63; int bc = i >> 6;
  float mx = -1e30f;
  for (int s = 0; s < SSZ; ++s)
    mx = fmaxf(mx, emb[(long)(bc * SSZ + s) * 64 + q]);
  str[i] = fmaxf(mx, 0.0f);
}

// single-wave WMMA FC: logits[16,10] = str[16,16384] * wfc[10,16384]^T + bfc
__global__ void fc_wmma_kernel(const float* __restrict__ str,
                               const float* __restrict__ wfc,
                               const float* __restrict__ bfc,
                               float* __restrict__ logits) {
  const int K = 16384;
  int lane = threadIdx.x & 31;
  int nl = lane & 15, half = lane >> 4;
  const float* arow = str + nl * K;
  const float* wrow = wfc + (nl < 10 ? nl : 0) * K;
  float wsc = (nl < 10) ? 1.0f : 0.0f;
  v8f acc = {};
  for (int kk = 0; kk < (K >> 2); ++kk) {
    int ka = kk * 4 + half * 2;
    v2f av = *(const v2f*)(arow + ka);
    v2f bv = *(const v2f*)(wrow + ka);
    bv.x *= wsc; bv.y *= wsc;
    acc = __builtin_amdgcn_wmma_f32_16x16x4_f32(false, av, false, bv,
                                                (short)0, acc, false, false);
  }
  for (int v = 0; v < 8; ++v) {
    int row = v + half * 8;
    int col = nl;
    if (col < 10) logits[row * 10 + col] = acc[v] + bfc[col];
  }
}

// ---------------------------------------------------------------------------
// host-side launch
// ---------------------------------------------------------------------------
extern "C" void kernel_launch(void* const* d_in, const int* in_sizes, int n_in,
                              void* d_out, int out_size, void* d_ws, size_t ws_size,
                              hipStream_t stream) {
  (void)in_sizes; (void)n_in; (void)out_size; (void)ws_size;
  const float* x   = (const float*)d_in[0];
  const float* w1  = (const float*)d_in[1];
  const float* b1  = (const float*)d_in[2];
  const float* g1  = (const float*)d_in[3];
  const float* be1 = (const float*)d_in[4];
  const float* w2  = (const float*)d_in[5];
  const float* b2  = (const float*)d_in[6];
  const float* wo1 = (const float*)d_in[7];
  const float* bo1 = (const float*)d_in[8];
  const float* wd1 = (const float*)d_in[9];
  const float* bd1 = (const float*)d_in[10];
  const float* g2  = (const float*)d_in[11];
  const float* be2 = (const float*)d_in[12];
  const float* wo2 = (const float*)d_in[13];
  const float* bo2 = (const float*)d_in[14];
  const float* wd2 = (const float*)d_in[15];
  const float* bd2 = (const float*)d_in[16];
  const float* g3  = (const float*)d_in[17];
  const float* be3 = (const float*)d_in[18];
  const float* wfc = (const float*)d_in[19];
  const float* bfc = (const float*)d_in[20];

  float* logits = (float*)d_out;          // 16*10
  float* emb    = (float*)d_out + 160;    // (16,256,4,8,8)

  float* ws = (float*)d_ws;
  size_t o = 0;
  float* h1   = ws + o; o += (size_t)BSZ * 64 * 1024;
  float* h2   = ws + o; o += (size_t)BSZ * 128 * 1024;
  float* off1 = ws + o; o += (size_t)BSZ * 144 * 1024;
  float* h3   = ws + o; o += (size_t)BSZ * 128 * 1024;
  float* h4   = ws + o; o += (size_t)BSZ * 128 * 256;
  float* off2 = ws + o; o += (size_t)BSZ * 144 * 256;
  float* h5   = ws + o; o += (size_t)BSZ * 256 * 256;
  float* str  = ws + o; o += (size_t)BSZ * 256 * 64;
  float* meanb= ws + o; o += 256;
  float* invb = ws + o; o += 256;
  // transposed (tap-major, zero-padded) weights
  float* w1T  = ws + o; o += (size_t)64 * 9 * 16;
  float* w2T  = ws + o; o += (size_t)128 * 9 * 64;
  float* wo1T = ws + o; o += (size_t)144 * 9 * 128;
  float* wd1T = ws + o; o += (size_t)128 * 9 * 128;
  float* wo2T = ws + o; o += (size_t)144 * 9 * 128;
  float* wd2T = ws + o; o += (size_t)256 * 9 * 128;

  // weight transposes (deterministic; run every call)
  {
    int t1 = 64 * 9 * 16;
    wtrans_kernel<<<(t1 + 255) / 256, 256, 0, stream>>>(w1, w1T, 64, 3, 16);
    int t2 = 128 * 9 * 64;
    wtrans_kernel<<<(t2 + 255) / 256, 256, 0, stream>>>(w2, w2T, 128, 64, 64);
    int t3 = 144 * 9 * 128;
    wtrans_kernel<<<(t3 + 255) / 256, 256, 0, stream>>>(wo1, wo1T, 144, 128, 128);
    int t4 = 128 * 9 * 128;
    wtrans_kernel<<<(t4 + 255) / 256, 256, 0, stream>>>(wd1, wd1T, 128, 128, 128);
    int t5 = 144 * 9 * 128;
    wtrans_kernel<<<(t5 + 255) / 256, 256, 0, stream>>>(wo2, wo2T, 144, 128, 128);
    int t6 = 256 * 9 * 128;
    wtrans_kernel<<<(t6 + 255) / 256, 256, 0, stream>>>(wd2, wd2T, 256, 128, 128);
  }

  for (int s = 0; s < SSZ; ++s) {
    // conv1 3->64 + relu (input tensor (B,3,S,32,32))
    conv3x3_wmma_kernel<true><<<dim3(64, 1, BSZ), 128, 0, stream>>>(
        x + (size_t)s * 1024, w1T, b1, h1, 3, 16, 64, 32, 32,
        3 * SSZ * 1024, SSZ * 1024, 1);
    bn_stats_kernel<<<64, 256, 0, stream>>>(h1, meanb, invb, 64, 1024);
    {
      int total = BSZ * 64 * 1024;
      bn_apply_kernel<<<(total + 255) / 256, 256, 0, stream>>>(
          h1, meanb, invb, g1, be1, 63, 10, total, 1);
    }
    // conv2 64->128
    conv3x3_wmma_kernel<false><<<dim3(64, 2, BSZ), 128, 0, stream>>>(
        h1, w2T, b2, h2, 64, 64, 128, 32, 32, 64 * 1024, 1024, 0);
    // offset conv1 128->144
    conv3x3_wmma_kernel<false><<<dim3(64, 3, BSZ), 128, 0, stream>>>(
        h2, wo1T, bo1, off1, 128, 128, 144, 32, 32, 128 * 1024, 1024, 0);
    // deform conv1 128->128 (+relu, bn2 takes relu input)
    deform_wmma_kernel<<<dim3(64, 2, BSZ), 128, 0, stream>>>(
        h2, off1, wd1T, bd1, h3, 128, 128, 32, 32, 1);
    bn_stats_kernel<<<128, 256, 0, stream>>>(h3, meanb, invb, 128, 1024);
    {
      int total = BSZ * 128 * 1024;
      bn_apply_kernel<<<(total + 255) / 256, 256, 0, stream>>>(
          h3, meanb, invb, g2 + s, be2 + s, 127, 10, total, 0);
    }
    {
      int total = BSZ * 128 * 16 * 16;
      maxpool_kernel<<<(total + 255) / 256, 256, 0, stream>>>(h3, h4, 128, 32, 32, 4, 4);
    }
    // offset conv2 128->144 on 16x16
    conv3x3_wmma_kernel<false><<<dim3(16, 3, BSZ), 128, 0, stream>>>(
        h4, wo2T, bo2, off2, 128, 128, 144, 16, 16, 128 * 256, 256, 0);
    // deform conv2 128->256, no relu
    deform_wmma_kernel<<<dim3(16, 4, BSZ), 128, 0, stream>>>(
        h4, off2, wd2T, bd2, h5, 128, 256, 16, 16, 0);
    bn_stats_kernel<<<256, 256, 0, stream>>>(h5, meanb, invb, 256, 256);
    {
      int total = BSZ * 256 * 256;
      bn_apply_kernel<<<(total + 255) / 256, 256, 0, stream>>>(
          h5, meanb, invb, g3 + s, be3 + s, 255, 8, total, 0);
    }
    {
      int total = BSZ * 256 * 8 * 8;
      maxpool_emb_kernel<<<(total + 255) / 256, 256, 0, stream>>>(h5, emb, 256, 16, 16, s);
    }
  }
  {
    int total = BSZ * 256 * 64;
    strength_kernel<<<(total + 255) / 256, 256, 0, stream>>>(emb, str, 256);
  }
  fc_wmma_kernel<<<1, 32, 0, stream>>>(str, wfc, bfc, logits);
}